// Awareness_5540507812461
// MI455X (gfx1250) — compile-verified
//
#include <hip/hip_runtime.h>
#include <math.h>
#include <stdint.h>

#define B_TOTAL  4096
#define DIMS     1024
#define STRIDE   1028          // 1024 + 4: row stride in floats -> bank-conflict-free B fetches
#define TILE     16
#define NTILES   (B_TOTAL / TILE)
#define NTHREADS 1024
#define NWAVES   (NTHREADS / 32)
#define INF_F    3.4e38f

typedef __attribute__((ext_vector_type(2))) float v2f;
typedef __attribute__((ext_vector_type(8))) float v8f;

__device__ __forceinline__ v8f wmma_f32_4(v2f a, v2f b, v8f c) {
  // 8-arg pattern: (neg_a, A, neg_b, B, c_mod, C, reuse_a, reuse_b)
  return __builtin_amdgcn_wmma_f32_16x16x4_f32(false, a, false, b, (short)0, c,
                                               false, false);
}

// CDNA5 async tensor path: memory -> LDS without VGPR round-trip (ASYNCcnt).
__device__ __forceinline__ void async_copy_b128(unsigned lds_off, const float* gptr) {
  asm volatile("global_load_async_to_lds_b128 %0, %1, off"
               :: "v"(lds_off), "v"(gptr) : "memory");
}
__device__ __forceinline__ void wait_async0() {
  asm volatile("s_wait_asynccnt 0" ::: "memory");
}

__global__ __launch_bounds__(NTHREADS, 1)
void awareness_scan_kernel(const float* __restrict__ x,
                           const int* __restrict__ labels,
                           float* __restrict__ out) {
  __shared__ float xTile[TILE * STRIDE];    // ~64.3 KB: current 16 candidates (padded rows)
  __shared__ float xn2[TILE];               // candidate squared norms
  __shared__ int   refIdx[B_TOTAL];         // 16 KB: ref slot -> row of x
  __shared__ float refNorm2[B_TOTAL];       // 16 KB: ref squared norms
  __shared__ float intraD2[TILE * TILE];    // 1 KB: intra-tile distances^2
  __shared__ float redMin2[NWAVES * TILE];
  __shared__ float redMax2[NWAVES * TILE];
  __shared__ int   redArg[NWAVES * TILE];
  __shared__ float preMin2[TILE];
  __shared__ float preMax2[TILE];
  __shared__ int   preArg[TILE];
  __shared__ int   shNRefs;

  const int  tid  = threadIdx.x;
  const int  w    = tid >> 5;
  const int  lane = tid & 31;
  const int  m16  = lane & 15;
  const bool hi   = lane >= 16;

  // scan state, private to thread 0 (uniform decisions broadcast via shNRefs)
  float R = 1.0f, minD = INF_F, maxD = 0.0f;
  int   intraC[TILE];

  const unsigned xTileBase = (unsigned)(uintptr_t)(void*)xTile; // LDS byte offset

  if (tid == 0) shNRefs = 0;
  __syncthreads();

  for (int tile = 0; tile < NTILES; ++tile) {
    const int nStart = shNRefs;

    // ---- stage X tile into LDS via async global->LDS B128 (one per thread) ----
    const float* src = x + (size_t)tile * (TILE * DIMS);
    {
      int r = tid >> 6, c = tid & 63;                       // 16 rows x 64 chunks
      unsigned ldsOff = xTileBase + (unsigned)(r * (STRIDE * 4) + c * 16);
      async_copy_b128(ldsOff, src + r * DIMS + c * 4);
    }
    if (tile + 1 < NTILES)
      __builtin_prefetch(src + TILE * DIMS + tid * 16, 0, 0); // global_prefetch_b8
    if (tid < TILE) xn2[tid] = 0.0f;
    wait_async0();
    __syncthreads();

    // ---- candidate squared norms ||x_c||^2 (conflict-free strided partials) ----
    {
      int c = tid >> 6, part = tid & 63;
      const float* p = &xTile[c * STRIDE + part];
      float s = 0.f;
      #pragma unroll
      for (int k = 0; k < DIMS / 64; ++k) { float v = p[k * 64]; s += v * v; }
      atomicAdd(&xn2[c], s);
    }
    __syncthreads();

    // ---- intra-tile Gram G = X X^T via WMMA (wave 0); d2 = n2m + n2n - 2G ----
    if (w == 0) {
      // A(16x4) lane L: rows M=L%16, K=k0+2*hi+{0,1};  B(4x16) lane L: col N=L%16,
      // same K pattern -> identical per-lane fetch, one ds_load feeds both.
      const float* gBase = &xTile[m16 * STRIDE + (hi ? 2 : 0)];
      v8f g = {0.f, 0.f, 0.f, 0.f, 0.f, 0.f, 0.f, 0.f};
      for (int k = 0; k < DIMS; k += 4) {
        v2f ab = *(const v2f*)(gBase + k);
        g = wmma_f32_4(ab, ab, g);
      }
      float myn2 = xn2[m16];
      #pragma unroll
      for (int v = 0; v < 8; ++v) {
        int m = v + (hi ? 8 : 0);
        intraD2[m * TILE + m16] = xn2[m] + myn2 - 2.0f * g[v];
      }
    }

    // ---- main GEMM: refs[0..nStart) x 16 candidates, fused min/argmin/max ----
    float bestM2 = INF_F, bestMax = -INF_F;
    int   bestJ = 0x7fffffff;
    const int   nRowTiles = (nStart + TILE - 1) >> 4;
    const float myXn2 = xn2[m16];
    for (int rt = w; rt < nRowTiles; rt += NWAVES) {
      int jrow = rt * TILE + m16;
      int gidx = (jrow < nStart) ? refIdx[jrow] : 0;
      const float* aBase = x + (size_t)gidx * DIMS + (hi ? 2 : 0); // L2-resident
      const float* bBase = &xTile[m16 * STRIDE + (hi ? 2 : 0)];
      v8f acc = {0.f, 0.f, 0.f, 0.f, 0.f, 0.f, 0.f, 0.f};
      for (int k = 0; k < DIMS; k += 4) {
        v2f a = *(const v2f*)(aBase + k);   // global_load_b64
        v2f b = *(const v2f*)(bBase + k);   // ds_load_b64 (conflict-free)
        acc = wmma_f32_4(a, b, acc);
      }
      #pragma unroll
      for (int v = 0; v < 8; ++v) {
        int j = rt * TILE + v + (hi ? 8 : 0);   // D layout: M = v + 8*laneHi
        if (j < nStart) {
          float d2 = refNorm2[j] - 2.0f * acc[v] + myXn2;
          if (d2 < bestM2) { bestM2 = d2; bestJ = j; }
          if (d2 > bestMax) bestMax = d2;
        }
      }
    }
    // combine lane halves (column n lives in lanes n and n+16)
    {
      float oM2 = __shfl_xor(bestM2, 16, 32);
      int   oJ  = __shfl_xor(bestJ, 16, 32);
      float oMx = __shfl_xor(bestMax, 16, 32);
      if (oM2 < bestM2 || (oM2 == bestM2 && oJ < bestJ)) { bestM2 = oM2; bestJ = oJ; }
      bestMax = fmaxf(bestMax, oMx);
    }
    if (lane < TILE) {
      redMin2[w * TILE + lane] = bestM2;
      redArg [w * TILE + lane] = bestJ;
      redMax2[w * TILE + lane] = bestMax;
    }
    __syncthreads();

    // ---- cross-wave reduction per candidate (first-min tie-break on j) ----
    if (tid < TILE) {
      float m2 = INF_F, mx = -INF_F;
      int   aj = 0x7fffffff;
      for (int ww = 0; ww < NWAVES; ++ww) {
        float om = redMin2[ww * TILE + tid];
        int   oj = redArg [ww * TILE + tid];
        if (om < m2 || (om == m2 && oj < aj)) { m2 = om; aj = oj; }
        mx = fmaxf(mx, redMax2[ww * TILE + tid]);
      }
      preMin2[tid] = m2; preArg[tid] = aj; preMax2[tid] = mx;
    }
    __syncthreads();

    // ---- sequential scan tail: 16 steps, thread 0 ----
    if (tid == 0) {
      int n = nStart, nIntra = 0;
      for (int c = 0; c < TILE; ++c) {
        int   gi  = tile * TILE + c;
        float m2  = preMin2[c], mx2 = preMax2[c];
        int   aj  = preArg[c];
        for (int t = 0; t < nIntra; ++t) {         // refs inserted this tile
          float dd = intraD2[intraC[t] * TILE + c];
          if (dd < m2) { m2 = dd; aj = nStart + t; }
          if (dd > mx2) mx2 = dd;
        }
        bool  anyRef  = n > 0;
        float minAct  = anyRef ? sqrtf(fmaxf(m2, 0.f)) : INF_F;
        float maxAct  = anyRef ? sqrtf(fmaxf(mx2, 0.f)) : -INF_F;
        bool  isFirst = (gi == 0);
        bool  doInsert = isFirst || (minAct > R);
        float pred;
        if (doInsert) {
          pred = (float)labels[gi];               // inserted ref is nearest (d=0)
          refIdx[n]   = gi;
          refNorm2[n] = xn2[c];
          intraC[nIntra] = c;
          ++nIntra; ++n;
        } else {
          int refGi = (aj >= nStart) ? (tile * TILE + intraC[aj - nStart])
                                     : refIdx[aj];
          pred = (float)labels[refGi];
        }
        if (!isFirst) {
          minD = fminf(minD, minAct);
          maxD = fmaxf(maxD, maxAct);
          R    = (minD + maxD) / 3.0f;
        }
        out[gi] = pred;
      }
      shNRefs = n;
    }
    __syncthreads();
  }
}

extern "C" void kernel_launch(void* const* d_in, const int* in_sizes, int n_in,
                              void* d_out, int out_size, void* d_ws, size_t ws_size,
                              hipStream_t stream) {
  (void)in_sizes; (void)n_in; (void)d_ws; (void)ws_size; (void)out_size;
  const float* x      = (const float*)d_in[0];
  const int*   labels = (const int*)d_in[1];
  float*       out    = (float*)d_out;
  awareness_scan_kernel<<<1, NTHREADS, 0, stream>>>(x, labels, out);
}